// MetaJanusRRPRAMAttn_50440095924319
// MI455X (gfx1250) — compile-verified
//
#include <hip/hip_runtime.h>
#include <math.h>

#define Bc 2
#define Tc 2048
#define Ec 1024
#define Hc 16
#define Dc 64
#define HDc (Hc*Dc)

typedef float v2f __attribute__((ext_vector_type(2)));
typedef float v8f __attribute__((ext_vector_type(8)));

// Async global->LDS copy of 16 bytes (per-lane), tracked by ASYNCcnt.
// VDST = LDS byte address (low 32 bits of flat shared pointer), VADDR = 64-bit
// global address. Bypasses VGPRs entirely (CDNA5 async-tensor path).
__device__ __forceinline__ void async_copy16(const float* g, const float* l)
{
    unsigned lds = (unsigned)(unsigned long long)l;   // LDS byte offset
    asm volatile("global_load_async_to_lds_b128 %0, %1, off"
                 :: "v"(lds), "v"(g) : "memory");
}
__device__ __forceinline__ void wait_async0()
{
    asm volatile("s_wait_asynccnt 0x0" ::: "memory");
}

// ---------------------------------------------------------------------------
// Generic batched WMMA fp32 GEMM, double-buffered async global->LDS staging.
//   BT = true :  C[M,N] = A[M,K] * B[N,K]^T   (NT — row-major weight "linear")
//   BT = false:  C[M,N] = A[M,K] * B[K,N]     (NN)
// M,N multiples of 64; K multiple of 16. Uses V_WMMA_F32_16X16X4_F32.
// Block = 128 threads = 4 waves, 64x64 tile, each wave owns a 32x32 subtile.
// ---------------------------------------------------------------------------
template<bool BT>
__global__ __launch_bounds__(128)
void gemm_f32_wmma(const float* __restrict__ A, const float* __restrict__ Bm,
                   float* __restrict__ C, int M, int N, int K,
                   long long sAb, long long sBb, long long sCb)
{
    const long long bz = blockIdx.z;
    A  += bz * sAb;  Bm += bz * sBb;  C += bz * sCb;

    // A tile [m][k] stride 20 floats (80B = 5*16B -> every 16B chunk aligned).
    // B tile: BT -> [n][k] stride 20 ; NN -> [k][n] stride 68 (272B = 17*16B).
    __shared__ __align__(16) float As[2][64][20];
    __shared__ __align__(16) float Bb[2][1280];       // 64*20 >= 16*68

    const int tid  = threadIdx.x;
    const int lane = tid & 31;
    const int wid  = tid >> 5;     // 0..3
    const int wr   = wid >> 1;     // wave row 0..1
    const int wc   = wid & 1;      // wave col 0..1
    const int r16  = lane & 15;
    const int hi   = lane >> 4;    // 0/1

    const int m0 = blockIdx.y * 64;
    const int n0 = blockIdx.x * 64;

    // issue the async copies for k-slab starting at k0n into buffer nb
    auto issue = [&](int k0n, int nb) {
        #pragma unroll
        for (int i = 0; i < 2; ++i) {                 // A: 64 rows x 64B
            const int chunk = tid * 2 + i;            // 0..255
            const int row = chunk >> 2, c4 = (chunk & 3) * 4;
            async_copy16(A + (long long)(m0 + row) * K + k0n + c4,
                         &As[nb][row][c4]);
        }
        if (BT) {                                     // B[N,K]: 64 n-rows x 64B
            #pragma unroll
            for (int i = 0; i < 2; ++i) {
                const int chunk = tid * 2 + i;
                const int row = chunk >> 2, c4 = (chunk & 3) * 4;
                async_copy16(Bm + (long long)(n0 + row) * K + k0n + c4,
                             &Bb[nb][row * 20 + c4]);
            }
        } else {                                      // B[K,N]: 16 k-rows x 256B
            #pragma unroll
            for (int i = 0; i < 2; ++i) {
                const int chunk = tid * 2 + i;
                const int kk = chunk >> 4, c4 = (chunk & 15) * 4;
                async_copy16(Bm + (long long)(k0n + kk) * N + n0 + c4,
                             &Bb[nb][kk * 68 + c4]);
            }
        }
    };

    v8f acc[2][2];
    acc[0][0] = {}; acc[0][1] = {}; acc[1][0] = {}; acc[1][1] = {};

    issue(0, 0);
    wait_async0();
    __syncthreads();

    int buf = 0;
    for (int k0 = 0; k0 < K; k0 += 16) {
        const int nbuf = buf ^ 1;
        if (k0 + 16 < K) issue(k0 + 16, nbuf);        // overlap copy with WMMA

        const float* Bp = &Bb[buf][0];
        #pragma unroll
        for (int kk = 0; kk < 16; kk += 4) {
            const int kb = kk + 2 * hi;               // lane's K pair
            v2f af[2], bf[2];
            #pragma unroll
            for (int i = 0; i < 2; ++i) {
                const int mr = wr * 32 + i * 16 + r16;
                af[i].x = As[buf][mr][kb];
                af[i].y = As[buf][mr][kb + 1];
            }
            #pragma unroll
            for (int j = 0; j < 2; ++j) {
                const int nc = wc * 32 + j * 16 + r16;
                if (BT) { bf[j].x = Bp[nc * 20 + kb];
                          bf[j].y = Bp[nc * 20 + kb + 1]; }
                else    { bf[j].x = Bp[kb * 68 + nc];
                          bf[j].y = Bp[(kb + 1) * 68 + nc]; }
            }
            #pragma unroll
            for (int i = 0; i < 2; ++i)
                #pragma unroll
                for (int j = 0; j < 2; ++j)
                    acc[i][j] = __builtin_amdgcn_wmma_f32_16x16x4_f32(
                        false, af[i], false, bf[j], (short)0, acc[i][j],
                        false, false);
        }

        wait_async0();          // next-stage copies landed in LDS
        __syncthreads();        // all waves done reading `buf` + see new data
        buf = nbuf;
    }

    // --- store: VGPR v holds row (v + 8*hi), col r16 of each 16x16 tile ---
    #pragma unroll
    for (int i = 0; i < 2; ++i)
        #pragma unroll
        for (int j = 0; j < 2; ++j) {
            const int col = n0 + wc * 32 + j * 16 + r16;
            #pragma unroll
            for (int v = 0; v < 8; ++v) {
                const int row = m0 + wr * 32 + i * 16 + v + 8 * hi;
                C[(long long)row * N + col] = acc[i][j][v];
            }
        }
}

// ---------------------------------------------------------------------------
// ra[b,h,t] = (1/8) * sum_e x[b,t,e] * wr[h,e,t]        (wr is [H,E,T])
// ---------------------------------------------------------------------------
__global__ void ra_kernel(const float* __restrict__ x, const float* __restrict__ wr,
                          float* __restrict__ ra)
{
    const int bh = blockIdx.x;                    // b*H + h
    const int b  = bh / Hc, h = bh % Hc;
    const int t  = blockIdx.y * 256 + threadIdx.x;
    const float* xr = x  + ((long long)b * Tc + t) * Ec;
    const float* wc = wr + (long long)h * Ec * Tc + t;
    float s = 0.f;
    for (int e = 0; e < Ec; ++e) s += xr[e] * wc[(long long)e * Tc];
    ra[(long long)bh * Tc + t] = s * 0.125f;      // 1/sqrt(D)
}

// ---------------------------------------------------------------------------
// RRP prefix-softmax scan: scores are per-key only -> cumulative softmax.
// One block per (b,h), 64 threads = one d-lane each, sequential over t.
// ---------------------------------------------------------------------------
__global__ __launch_bounds__(64)
void rrp_scan_kernel(const float* __restrict__ ra, const float* __restrict__ rv,
                     float* __restrict__ rrp)
{
    const int bh = blockIdx.x;
    const int b  = bh / Hc, h = bh % Hc;
    const int tid = threadIdx.x;                  // d = tid (0..63)
    const float* raw = ra + (long long)bh * Tc;

    __shared__ float red[64];
    float m = -1e30f;
    for (int t = tid; t < Tc; t += 64) m = fmaxf(m, raw[t]);
    red[tid] = m; __syncthreads();
    for (int off = 32; off > 0; off >>= 1) {
        if (tid < off) red[tid] = fmaxf(red[tid], red[tid + off]);
        __syncthreads();
    }
    const float M = red[0];

    const float* rvp = rv  + (long long)b * Tc * HDc + h * Dc + tid;
    float*       op  = rrp + (long long)b * Tc * HDc + h * Dc + tid;
    float z = 0.f, acc = 0.f;
    for (int t = 0; t < Tc; ++t) {
        const float e = __expf(raw[t] - M);
        z   += e;
        acc += e * rvp[(long long)t * HDc];
        op[(long long)t * HDc] = acc / z;
    }
}

// ---------------------------------------------------------------------------
// sc2[b,t] = dot(x[b,t,:], eb[b,t,:]) / sqrt(E)
// ---------------------------------------------------------------------------
__global__ __launch_bounds__(256)
void rowdot_kernel(const float* __restrict__ x, const float* __restrict__ eb,
                   float* __restrict__ sc2)
{
    const long long r = blockIdx.x;               // b*T + t
    __shared__ float red[256];
    float s = 0.f;
    for (int e = threadIdx.x; e < Ec; e += 256)
        s += x[r * Ec + e] * eb[r * Ec + e];
    red[threadIdx.x] = s; __syncthreads();
    for (int off = 128; off > 0; off >>= 1) {
        if (threadIdx.x < off) red[threadIdx.x] += red[threadIdx.x + off];
        __syncthreads();
    }
    if (threadIdx.x == 0) sc2[r] = red[0] * (1.0f / 32.0f);  // 1/sqrt(1024)
}

// ---------------------------------------------------------------------------
// Janus row stats: m_s = max_{t<=s} a_s*c_t ; invZ_s = 1/sum exp(a_s*c_t - m_s)
// sc2 row staged in LDS (read ~s times per thread).
// ---------------------------------------------------------------------------
__global__ __launch_bounds__(256)
void janus_mz_kernel(const float* __restrict__ sc2,
                     float* __restrict__ mbuf, float* __restrict__ zbuf)
{
    const int b = blockIdx.x;
    const int s = blockIdx.y * 256 + threadIdx.x;
    const float* c = sc2 + (long long)b * Tc;

    __shared__ float cs[Tc];
    for (int t = threadIdx.x; t < Tc; t += 256) cs[t] = c[t];
    __syncthreads();

    const float a = cs[s];
    float m = -1e30f;
    for (int t = 0; t <= s; ++t) m = fmaxf(m, a * cs[t]);
    float Z = 0.f;
    for (int t = 0; t <= s; ++t) Z += __expf(a * cs[t] - m);
    mbuf[(long long)b * Tc + s] = m;
    zbuf[(long long)b * Tc + s] = 1.0f / Z;
}

// ---------------------------------------------------------------------------
// Materialize P[b,s,t] = (t<=s) ? exp(a_s*c_t - m_s)*invZ_s : 0
// ---------------------------------------------------------------------------
__global__ __launch_bounds__(256)
void janus_p_kernel(const float* __restrict__ sc2, const float* __restrict__ mbuf,
                    const float* __restrict__ zbuf, float* __restrict__ P)
{
    const long long bs = blockIdx.x;              // b*T + s
    const int b = (int)(bs / Tc), s = (int)(bs % Tc);
    const float a = sc2[bs], m = mbuf[bs], iz = zbuf[bs];
    const float* c = sc2 + (long long)b * Tc;
    float* Pr = P + bs * Tc;
    for (int t = threadIdx.x; t < Tc; t += 256)
        Pr[t] = (t <= s) ? __expf(a * c[t] - m) * iz : 0.f;
}

// ---------------------------------------------------------------------------
// fused = softmax(gate)[h,0]*rrp + softmax(gate)[h,1]*jan
// ---------------------------------------------------------------------------
__global__ __launch_bounds__(256)
void fuse_kernel(const float* __restrict__ rrp, const float* __restrict__ jan,
                 const float* __restrict__ gate, float* __restrict__ fused)
{
    const long long idx = (long long)blockIdx.x * 256 + threadIdx.x;
    if (idx >= (long long)Bc * Tc * HDc) return;
    const int h = (int)((idx % HDc) / Dc);
    const float g0 = gate[h * 2 + 0], g1 = gate[h * 2 + 1];
    const float mg = fmaxf(g0, g1);
    const float e0 = __expf(g0 - mg), e1 = __expf(g1 - mg);
    const float inv = 1.0f / (e0 + e1);
    fused[idx] = (e0 * inv) * rrp[idx] + (e1 * inv) * jan[idx];
}

// ---------------------------------------------------------------------------
extern "C" void kernel_launch(void* const* d_in, const int* in_sizes, int n_in,
                              void* d_out, int out_size, void* d_ws, size_t ws_size,
                              hipStream_t stream)
{
    const float* x     = (const float*)d_in[0];   // [B,T,E]
    const float* wr    = (const float*)d_in[1];   // [H,E,T]
    const float* wvr_w = (const float*)d_in[2];   // [H*D,E]
    const float* wj_w  = (const float*)d_in[3];   // [E,E]
    const float* gate  = (const float*)d_in[4];   // [H,2]
    const float* wo_w  = (const float*)d_in[5];   // [E,H*D]
    float* out = (float*)d_out;                   // [B,T,E]

    const long long BT   = (long long)Bc * Tc;        // 4096
    const long long BTHD = BT * HDc;                  // 4,194,304 floats

    // workspace layout (floats), ~135 MB total
    float* ws    = (float*)d_ws;
    float* rv    = ws;                    // BTHD
    float* echo  = rv   + BTHD;           // BTHD
    float* eb    = echo + BTHD;           // BTHD
    float* ra    = eb   + BTHD;           // B*H*T
    float* rrp   = ra   + (long long)Bc * Hc * Tc;    // BTHD
    float* sc2   = rrp  + BTHD;           // B*T
    float* mbuf  = sc2  + BT;             // B*T
    float* zbuf  = mbuf + BT;             // B*T
    float* P     = zbuf + BT;             // B*T*T
    float* jan   = P    + (long long)Bc * Tc * Tc;    // BTHD
    float* fused = jan  + BTHD;           // BTHD

    dim3 blk(128);

    // 1) rv = x @ wvr_w^T        [BT x HD], K=E   (NT)
    gemm_f32_wmma<true><<<dim3(HDc / 64, (int)BT / 64, 1), blk, 0, stream>>>(
        x, wvr_w, rv, (int)BT, HDc, Ec, 0, 0, 0);

    // 2) echo = x @ wj_w^T       [BT x E],  K=E   (NT)
    gemm_f32_wmma<true><<<dim3(Ec / 64, (int)BT / 64, 1), blk, 0, stream>>>(
        x, wj_w, echo, (int)BT, Ec, Ec, 0, 0, 0);

    // 3) eb = echo @ wj_w        [BT x E],  K=E   (NN)
    gemm_f32_wmma<false><<<dim3(Ec / 64, (int)BT / 64, 1), blk, 0, stream>>>(
        echo, wj_w, eb, (int)BT, Ec, Ec, 0, 0, 0);

    // 4) ra scores
    ra_kernel<<<dim3(Bc * Hc, Tc / 256), dim3(256), 0, stream>>>(x, wr, ra);

    // 5) RRP cumulative-softmax scan
    rrp_scan_kernel<<<dim3(Bc * Hc), dim3(64), 0, stream>>>(ra, rv, rrp);

    // 6) sc2 row dots
    rowdot_kernel<<<dim3((int)BT), dim3(256), 0, stream>>>(x, eb, sc2);

    // 7) Janus per-row max / normalizer
    janus_mz_kernel<<<dim3(Bc, Tc / 256), dim3(256), 0, stream>>>(sc2, mbuf, zbuf);

    // 8) Materialize P
    janus_p_kernel<<<dim3((int)BT), dim3(256), 0, stream>>>(sc2, mbuf, zbuf, P);

    // 9) jan = P @ echo  (batched NN, per b: [T,T] x [T,E])
    gemm_f32_wmma<false><<<dim3(Ec / 64, Tc / 64, Bc), blk, 0, stream>>>(
        P, echo, jan, Tc, Ec, Tc,
        (long long)Tc * Tc, (long long)Tc * Ec, (long long)Tc * HDc);

    // 10) gated fusion
    fuse_kernel<<<dim3((int)((BTHD + 255) / 256)), dim3(256), 0, stream>>>(
        rrp, jan, gate, fused);

    // 11) out = fused @ wo_w^T   [BT x E], K=HD  (NT)
    gemm_f32_wmma<true><<<dim3(Ec / 64, (int)BT / 64, 1), blk, 0, stream>>>(
        fused, wo_w, out, (int)BT, Ec, HDc, 0, 0, 0);
}